// SparseIndexer_755914244590
// MI455X (gfx1250) — compile-verified
//
#include <hip/hip_runtime.h>
#include <stdint.h>

// ---------------------------------------------------------------------------
// SparseIndexer: indices = top64( Q @ K^T ) per (head, q-row).
// MI455X / gfx1250, wave32. Scores via V_WMMA_F32_16X16X4_F32 (exact f32 —
// matmul is NOT the bottleneck, selection is, so no precision tradeoff).
// Positive scaling by 1/sqrt(d) does not change top-k indices -> dropped.
// Round-2 change: 4 independent WMMA accumulator chains per tile to break
// the serial D->C RAW dependency visible in the round-1 assembly (only
// 2 waves/SIMD at 262KB LDS, so ILP must come from within the wave).
// ---------------------------------------------------------------------------

typedef __attribute__((ext_vector_type(2))) float v2f;
typedef __attribute__((ext_vector_type(8))) float v8f;

#define HEADS   16
#define SEQ     4096
#define DHEAD   128
#define TOPK    64
#define QTILE   16                    // q rows per workgroup
#define WAVES   8
#define THREADS (WAVES * 32)
#define KTILES      (SEQ / 16)        // 256
#define KT_PER_WAVE (KTILES / WAVES)  // 32
#define NCHUNK  (DHEAD / 4)           // 32 K-chunks of 4 for f32 WMMA
#define NACC    4                     // independent accumulator chains
// LDS score block: 16 rows x 4100 floats. Stride 4100 (== 4 mod 64 banks)
// makes the C-tile scatter stores conflict-free. 262,400 B < 320 KB WGP LDS.
#define PADW    4100
#define LDS_BYTES (QTILE * PADW * 4)

// Monotonic float->uint mapping (total order matches float <).
__device__ __forceinline__ unsigned int f32_sortable(float f) {
  unsigned int u = __float_as_uint(f);
  return (u & 0x80000000u) ? ~u : (u | 0x80000000u);
}

// 5-step wave32 max reduction on 64-bit keys.
__device__ __forceinline__ unsigned long long wave_max_u64(unsigned long long k) {
#pragma unroll
  for (int off = 16; off > 0; off >>= 1) {
    unsigned long long o = __shfl_xor(k, off, 32);
    if (o > k) k = o;
  }
  return k;
}

// Insert key into descending sorted 4-register cache (c0 >= c1 >= c2 >= c3).
__device__ __forceinline__ void ins4(unsigned long long key,
                                     unsigned long long& c0, unsigned long long& c1,
                                     unsigned long long& c2, unsigned long long& c3) {
  if (key > c3) {
    if (key > c2) {
      c3 = c2;
      if (key > c1) {
        c2 = c1;
        if (key > c0) { c1 = c0; c0 = key; } else { c1 = key; }
      } else {
        c2 = key;
      }
    } else {
      c3 = key;
    }
  }
}

__global__ void __launch_bounds__(THREADS, 1)
sparse_indexer_topk(const float* __restrict__ q,
                    const float* __restrict__ kmat,
                    int* __restrict__ out) {
  extern __shared__ float smem[];           // [QTILE][PADW] f32 scores
  const int qt   = blockIdx.x;              // 0..255  (q tile)
  const int h    = blockIdx.y;              // 0..15   (head)
  const int lane = threadIdx.x & 31;
  const int wave = threadIdx.x >> 5;

  // ---------------- A fragment: Q tile 16x128 (f32 16x4 WMMA A layout) -----
  // lanes 0-15: M=lane, K pair {0,1} of chunk; lanes 16-31: M=lane-16, {2,3}.
  const float* qbase = q + (size_t)(h * SEQ + qt * QTILE) * DHEAD;
  const int am = lane & 15;
  const int ak = (lane >> 4) * 2;
  v2f a_frag[NCHUNK];
#pragma unroll
  for (int c = 0; c < NCHUNK; ++c) {
    const float* p = qbase + am * DHEAD + 4 * c + ak;
    a_frag[c] = (v2f){p[0], p[1]};
  }

  // ---------------- score matmul: wave owns KT_PER_WAVE tiles of 16 k-rows -
  // B layout (4x16 f32): lane n=lane&15 holds column n; VGPR pair holds
  // K={0,1} (lanes 0-15) / K={2,3} (lanes 16-31) of the chunk. Since
  // B[k][n] = K[n][d], this is a contiguous float2 read of K's row n.
  const int bn = lane & 15;
  const int bk = (lane >> 4) * 2;
  for (int i = 0; i < KT_PER_WAVE; ++i) {
    const int kt = wave * KT_PER_WAVE + i;
    const float* kbase =
        kmat + (size_t)(h * SEQ + kt * 16 + bn) * DHEAD + bk;
    if (i + 1 < KT_PER_WAVE)
      __builtin_prefetch(kbase + 16 * DHEAD, 0, 0);  // global_prefetch_b8

    // 4 independent accumulator chains: WMMA c is dependent only on WMMA
    // c-4, so the scheduler can keep the matrix pipe busy within one wave.
    v8f acc[NACC];
#pragma unroll
    for (int a = 0; a < NACC; ++a)
      acc[a] = (v8f){0.f, 0.f, 0.f, 0.f, 0.f, 0.f, 0.f, 0.f};

#pragma unroll
    for (int c = 0; c < NCHUNK; ++c) {
      const float* p = kbase + 4 * c;
      v2f b = (v2f){p[0], p[1]};
      acc[c & (NACC - 1)] = __builtin_amdgcn_wmma_f32_16x16x4_f32(
          /*neg_a=*/false, a_frag[c], /*neg_b=*/false, b,
          /*c_mod=*/(short)0, acc[c & (NACC - 1)],
          /*reuse_a=*/false, /*reuse_b=*/false);
    }

    // C layout: VGPR r -> row r (lanes 0-15) / row 8+r (lanes 16-31), col=lane&15.
    const int colb   = kt * 16 + (lane & 15);
    const int rshift = (lane >> 4) * 8;
#pragma unroll
    for (int r = 0; r < 8; ++r)
      smem[(r + rshift) * PADW + colb] =
          (acc[0][r] + acc[1][r]) + (acc[2][r] + acc[3][r]);
  }
  __syncthreads();

  // ---------------- top-64 selection: 2 rows per wave ----------------------
  // key = (sortable(score) << 32) | (4095 - col): unique, sorts by value
  // descending then index ascending (matches lax.top_k stable tie order).
  // Per-lane top-4 register cache over its 128 strided elements; extracted
  // elements are overwritten with -inf in LDS (-inf keys sort below every
  // finite key, so no skip logic is needed on rebuild). Only the owner lane
  // touches its own stride (col % 32 == lane), so same-thread LDS ordering
  // suffices.
  for (int rr = 0; rr < 2; ++rr) {
    const int row = wave * 2 + rr;
    float* srow = smem + row * PADW;

    unsigned long long c0 = 0, c1 = 0, c2 = 0, c3 = 0;
    for (int i = lane; i < SEQ; i += 32) {
      unsigned long long key =
          ((unsigned long long)f32_sortable(srow[i]) << 32) |
          (unsigned int)(SEQ - 1 - i);
      ins4(key, c0, c1, c2, c3);
    }

    int out0 = 0, out1 = 0;
    for (int t = 0; t < TOPK; ++t) {
      const unsigned long long m = wave_max_u64(c0);
      const int idx = (SEQ - 1) - (int)(unsigned int)m;   // low 32 bits
      if (c0 == m) {                                      // unique owner
        srow[idx] = -__builtin_huge_valf();               // consume
        c0 = c1; c1 = c2; c2 = c3; c3 = 0;
        if (c0 == 0ull) {                                 // cache empty: rebuild
          for (int i = lane; i < SEQ; i += 32) {
            unsigned long long key =
                ((unsigned long long)f32_sortable(srow[i]) << 32) |
                (unsigned int)(SEQ - 1 - i);
            ins4(key, c0, c1, c2, c3);
          }
        }
      }
      if ((t & 31) == lane) { if (t < 32) out0 = idx; else out1 = idx; }
    }

    int* orow = out + (size_t)(h * SEQ + qt * QTILE + row) * TOPK;
    orow[lane]      = out0;   // coalesced: t = 0..31
    orow[32 + lane] = out1;   // coalesced: t = 32..63
  }
}

// ---------------------------------------------------------------------------
extern "C" void kernel_launch(void* const* d_in, const int* in_sizes, int n_in,
                              void* d_out, int out_size, void* d_ws, size_t ws_size,
                              hipStream_t stream) {
  (void)in_sizes; (void)n_in; (void)out_size; (void)d_ws; (void)ws_size;
  const float* q = (const float*)d_in[0];
  const float* k = (const float*)d_in[1];
  int* out = (int*)d_out;

  // 262,400 B dynamic LDS per workgroup (gfx1250 allows up to 320 KB/WGP).
  hipFuncSetAttribute((const void*)sparse_indexer_topk,
                      hipFuncAttributeMaxDynamicSharedMemorySize, LDS_BYTES);

  dim3 grid(SEQ / QTILE, HEADS);     // 256 x 16 workgroups
  sparse_indexer_topk<<<grid, THREADS, LDS_BYTES, stream>>>(q, k, out);
}